// Block_50422916055264
// MI455X (gfx1250) — compile-verified
//
#include <hip/hip_runtime.h>

typedef __attribute__((ext_vector_type(2))) float v2f;
typedef __attribute__((ext_vector_type(8))) float v8f;
typedef __attribute__((ext_vector_type(4))) unsigned int u32x4;
typedef __attribute__((ext_vector_type(8))) unsigned int u32x8;

#define N_SITES 400000
#define R_PAIRS 200000
#define K_OFF   27
#define N_IN    64
#define N_OUT   64
#define BN_EPS  1e-4f
#define LEAK    (1.0f / 3.0f)

#define TILES_PER_K (R_PAIRS / 16)   // 12500 (R divisible by 16)

// ---------------------------------------------------------------------------
// Kernel 1: rulebook gather -> 16x64 @ 64x64 via v_wmma_f32_16x16x4_f32 -> scatter-add
// W[k] (16 KB contiguous) staged into LDS by the Tensor Data Mover (1 op/block).
// ---------------------------------------------------------------------------
__global__ __launch_bounds__(256) void conv_wmma(const float* __restrict__ feat,
                                                 const float* __restrict__ W,
                                                 const int* __restrict__ in_idx,
                                                 const int* __restrict__ out_idx,
                                                 float* __restrict__ out) {
    __shared__ float Wl[N_IN * N_OUT];  // 16 KB: W[k], row-major [in][out]

    const int k = blockIdx.y;
    const float* Wk = W + (size_t)k * N_IN * N_OUT;

    // --- TDM stage: one tensor_load_to_lds issued by wave 0 (EXEC-independent DMA)
    if (threadIdx.x < 32) {
        const unsigned long long ga = (unsigned long long)(const void*)Wk;
        const unsigned lds_off = (unsigned)(unsigned long long)(void*)&Wl[0];

        u32x4 g0;
        g0.x = 1u;                                   // count=1, is_restore=0, gather off
        g0.y = lds_off;                              // lds_addr
        g0.z = (unsigned)ga;                         // global_addr[31:0]
        g0.w = (unsigned)(ga >> 32) | (2u << 30);    // global_addr[56:32] | type=2

        u32x8 g1;
        g1[0] = 2u << 16;                            // workgroup_mask=0, data_size=4B
        g1[1] = 4096u << 16;                         // tensor_dim0 = 4096 elements
        g1[2] = 0u;                                  // tensor_dim1 unused
        g1[3] = 4096u << 16;                         // tile_dim0 = 4096 (full tile)
        g1[4] = 0u;                                  // tile_dim1 = 0 -> 1-D tile
        g1[5] = 0u;                                  // strides unused for 1-D tile
        g1[6] = 0u;
        g1[7] = 0u;

        u32x4 g2 = {0u, 0u, 0u, 0u};                 // dims 2..4 unused
        u32x4 g3 = {0u, 0u, 0u, 0u};

        asm volatile("tensor_load_to_lds %0, %1, %2, %3"
                     :: "s"(g0), "s"(g1), "s"(g2), "s"(g3)
                     : "memory");
        __builtin_amdgcn_s_wait_tensorcnt(0);
    }
    __syncthreads();

    const int wave = threadIdx.x >> 5;
    const int lane = threadIdx.x & 31;
    const int lo   = lane & 15;     // M (A) / N (B,C) position
    const int hi   = lane >> 4;     // K-parity half of the wave

    const int tile = blockIdx.x * 8 + wave;   // wave-uniform guard -> EXEC all-ones
    if (tile >= TILES_PER_K) return;

    const int base = k * R_PAIRS + tile * 16;
    const int irow = in_idx[base + lo];   // lanes 16..31 duplicate lanes 0..15 (cached)
    const int orow = out_idx[base + lo];

    // A: 16x64 gathered rows in WMMA 16x4xf32 layout.
    // Instruction kk covers K = 4kk..4kk+3; lane half hi supplies K = 4kk+2hi, 4kk+2hi+1.
    const float* ap = feat + (size_t)irow * N_IN + 2 * hi;
    v2f A[16];
#pragma unroll
    for (int kk = 0; kk < 16; ++kk)
        A[kk] = *(const v2f*)(ap + 4 * kk);   // global_load_b64, 8B aligned

    v8f acc0 = {}, acc1 = {}, acc2 = {}, acc3 = {};

#pragma unroll
    for (int kk = 0; kk < 16; ++kk) {
        const int r0 = (4 * kk + 2 * hi) * N_OUT;   // B rows for this lane half
        const int r1 = r0 + N_OUT;
#pragma unroll
        for (int nt = 0; nt < 4; ++nt) {
            v2f b;
            b.x = Wl[r0 + nt * 16 + lo];
            b.y = Wl[r1 + nt * 16 + lo];
            v8f c = (nt == 0) ? acc0 : (nt == 1) ? acc1 : (nt == 2) ? acc2 : acc3;
            c = __builtin_amdgcn_wmma_f32_16x16x4_f32(false, A[kk], false, b,
                                                      (short)0, c, false, false);
            if      (nt == 0) acc0 = c;
            else if (nt == 1) acc1 = c;
            else if (nt == 2) acc2 = c;
            else              acc3 = c;
        }
    }

    // C/D layout: VGPR r holds M = r + 8*hi, N = 16*nt + lo. Scatter-add to L2.
#pragma unroll
    for (int r = 0; r < 8; ++r) {
        const int m    = r + 8 * hi;
        const int drow = __shfl(orow, m, 32);
        float* op = out + (size_t)drow * N_OUT + lo;
        __hip_atomic_fetch_add(op +  0, acc0[r], __ATOMIC_RELAXED, __HIP_MEMORY_SCOPE_AGENT);
        __hip_atomic_fetch_add(op + 16, acc1[r], __ATOMIC_RELAXED, __HIP_MEMORY_SCOPE_AGENT);
        __hip_atomic_fetch_add(op + 32, acc2[r], __ATOMIC_RELAXED, __HIP_MEMORY_SCOPE_AGENT);
        __hip_atomic_fetch_add(op + 48, acc3[r], __ATOMIC_RELAXED, __HIP_MEMORY_SCOPE_AGENT);
    }
}

// ---------------------------------------------------------------------------
// Kernel 2: per-channel sum / sum-of-squares into ws[0..63] / ws[64..127]
// ---------------------------------------------------------------------------
__global__ __launch_bounds__(256) void stats_kernel(const float* __restrict__ out,
                                                    float* __restrict__ ws) {
    const int c   = threadIdx.x & 63;
    const int grp = threadIdx.x >> 6;   // 4 row-streams per block
    float s = 0.f, s2 = 0.f;
    for (int row = blockIdx.x * 4 + grp; row < N_SITES; row += gridDim.x * 4) {
        float v = out[(size_t)row * N_OUT + c];
        s += v;
        s2 += v * v;
    }
    __shared__ float ls[4][64], ls2[4][64];
    ls[grp][c] = s;
    ls2[grp][c] = s2;
    __syncthreads();
    if (grp == 0) {
        s  = ls[0][c] + ls[1][c] + ls[2][c] + ls[3][c];
        s2 = ls2[0][c] + ls2[1][c] + ls2[2][c] + ls2[3][c];
        __hip_atomic_fetch_add(&ws[c],      s,  __ATOMIC_RELAXED, __HIP_MEMORY_SCOPE_AGENT);
        __hip_atomic_fetch_add(&ws[64 + c], s2, __ATOMIC_RELAXED, __HIP_MEMORY_SCOPE_AGENT);
    }
}

// ---------------------------------------------------------------------------
// Kernel 3: in-place BN (training stats) + LeakyReLU(1/3). Conv bias cancels in BN.
// ---------------------------------------------------------------------------
__global__ __launch_bounds__(256) void bn_lrelu(float* __restrict__ out,
                                                const float* __restrict__ ws,
                                                const float* __restrict__ bnw,
                                                const float* __restrict__ bnb) {
    const size_t idx = (size_t)blockIdx.x * 256 + threadIdx.x;
    if (idx >= (size_t)N_SITES * N_OUT) return;
    const int c = (int)(idx & 63);
    const float inv_n = 1.0f / (float)N_SITES;
    const float mean  = ws[c] * inv_n;
    const float var   = ws[64 + c] * inv_n - mean * mean;
    const float x = out[idx];
    float y = (x - mean) * rsqrtf(var + BN_EPS) * bnw[c] + bnb[c];
    out[idx] = (y >= 0.f) ? y : y * LEAK;
}

// ---------------------------------------------------------------------------
extern "C" void kernel_launch(void* const* d_in, const int* in_sizes, int n_in,
                              void* d_out, int out_size, void* d_ws, size_t ws_size,
                              hipStream_t stream) {
    const float* feat   = (const float*)d_in[0];
    const float* W      = (const float*)d_in[1];
    // d_in[2] = bias (cancels exactly through training-stat BN; skipped)
    const float* bnw    = (const float*)d_in[3];
    const float* bnb    = (const float*)d_in[4];
    const int*   in_idx  = (const int*)d_in[5];
    const int*   out_idx = (const int*)d_in[6];
    float* out = (float*)d_out;
    float* ws  = (float*)d_ws;

    // Zero accumulator + stats region (memset nodes are graph-capture safe).
    hipMemsetAsync(out, 0, (size_t)N_SITES * N_OUT * sizeof(float), stream);
    hipMemsetAsync(ws, 0, 128 * sizeof(float), stream);

    // 1) sparse conv: 8 waves/block, 1 tile(16 pairs)/wave, one k per blockIdx.y
    dim3 grid((TILES_PER_K + 7) / 8, K_OFF);
    conv_wmma<<<grid, 256, 0, stream>>>(feat, W, in_idx, out_idx, out);

    // 2) channel statistics
    stats_kernel<<<1024, 256, 0, stream>>>(out, ws);

    // 3) normalize + leaky relu
    const size_t total = (size_t)N_SITES * N_OUT;
    bn_lrelu<<<(unsigned)((total + 255) / 256), 256, 0, stream>>>(out, ws, bnw, bnb);
}